// MCKRL_9723805958732
// MI455X (gfx1250) — compile-verified
//
#include <hip/hip_runtime.h>
#include <hip/hip_bf16.h>
#include <math.h>

typedef __attribute__((ext_vector_type(2))) float v2f;
typedef __attribute__((ext_vector_type(8))) float v8f;

#define NODES 30000
#define NEDGE 250000
#define INF_  512
#define HIDF  256
#define OUTF  128

// ---------------- utility kernels ----------------

__global__ void k_zero(float* __restrict__ p, int n) {
    int t = blockIdx.x * blockDim.x + threadIdx.x;
    if (t < n) p[t] = 0.0f;
}

__global__ void k_zero_int(int* __restrict__ p, int n) {
    int t = blockIdx.x * blockDim.x + threadIdx.x;
    if (t < n) p[t] = 0;
}

// counts -> rsqrt(max(count,1))
__global__ void k_rsqrt_clamp(float* __restrict__ p, int n) {
    int t = blockIdx.x * blockDim.x + threadIdx.x;
    if (t < n) p[t] = rsqrtf(fmaxf(p[t], 1.0f));
}

// degree histograms (float counts; exact for small ints)
__global__ void k_count_deg(const int* __restrict__ src, const int* __restrict__ dst,
                            float* __restrict__ cnt_out, float* __restrict__ cnt_in, int E) {
    int t = blockIdx.x * blockDim.x + threadIdx.x;
    if (t >= E) return;
    atomicAdd(cnt_out + src[t], 1.0f);
    atomicAdd(cnt_in + dst[t], 1.0f);
}

// int histogram of dst for CSR build
__global__ void k_count_int(const int* __restrict__ dst, int* __restrict__ cnt, int E) {
    int t = blockIdx.x * blockDim.x + threadIdx.x;
    if (t >= E) return;
    atomicAdd(cnt + dst[t], 1);
}

// single-block exclusive prefix scan: cnt[n] -> row_ptr[n+1]
__global__ void k_scan(const int* __restrict__ cnt, int* __restrict__ row_ptr, int n) {
    __shared__ int part[1024];
    int t = threadIdx.x;
    int chunk = (n + 1023) / 1024;
    int beg = t * chunk;
    int end = min(beg + chunk, n);
    int s = 0;
    for (int i = beg; i < end; ++i) s += cnt[i];
    part[t] = s;
    __syncthreads();
    for (int off = 1; off < 1024; off <<= 1) {
        int v = (t >= off) ? part[t - off] : 0;
        __syncthreads();
        part[t] += v;
        __syncthreads();
    }
    int run = part[t] - s;                 // exclusive offset of this chunk
    for (int i = beg; i < end; ++i) { row_ptr[i] = run; run += cnt[i]; }
    if (t == 0) row_ptr[n] = part[1023];
}

// fill CSR: col_idx sorted by dst, stores src ids
__global__ void k_fill(const int* __restrict__ src, const int* __restrict__ dst,
                       const int* __restrict__ row_ptr, int* __restrict__ cursor,
                       int* __restrict__ col_idx, int E) {
    int t = blockIdx.x * blockDim.x + threadIdx.x;
    if (t >= E) return;
    int d = dst[t];
    int pos = atomicAdd(cursor + d, 1);
    col_idx[row_ptr[d] + pos] = src[t];
}

// buf[n*F+f] = bA[f] + bB[f]   (F is a power of 2)
__global__ void k_init_bias2(float* __restrict__ buf, const float* __restrict__ bA,
                             const float* __restrict__ bB, int fmask, int total) {
    int t = blockIdx.x * blockDim.x + threadIdx.x;
    if (t >= total) return;
    int f = t & fmask;
    buf[t] = bA[f] + bB[f];
}

__global__ void k_relu(float* __restrict__ p, int n) {
    int t = blockIdx.x * blockDim.x + threadIdx.x;
    if (t < n) p[t] = fmaxf(p[t], 0.0f);
}

// ---------------- WMMA fp32 GEMM:  Y[M,F] = X[M,K] @ W[K,F] ----------------
// One wave computes a 16x64 strip: 4 tiles of 16x16, A fragment reused.
__global__ void k_gemm_wmma_f32(const float* __restrict__ X, const float* __restrict__ W,
                                float* __restrict__ Y, int K, int F,
                                int row_tiles, int col_strips) {
    int wid  = (int)((blockIdx.x * blockDim.x + threadIdx.x) >> 5); // wave-uniform
    int lane = threadIdx.x & 31;
    int tasks = row_tiles * col_strips;
    if (wid >= tasks) return;                 // whole wave exits -> EXEC stays all-ones
    int rt = wid / col_strips;
    int cs = wid - rt * col_strips;
    int rowbase = rt << 4;
    int colbase = cs << 6;
    int m = lane & 15;
    int g = lane >> 4;

    const float* Xrow = X + (size_t)(rowbase + m) * (size_t)K;
    int col = colbase + m;

    v8f acc0 = {}, acc1 = {}, acc2 = {}, acc3 = {};

    for (int k = 0; k < K; k += 4) {
        v2f a = *(const v2f*)(Xrow + k + 2 * g);           // A[m][k+2g .. k+2g+1]
        const float* Wk = W + (size_t)(k + 2 * g) * (size_t)F;
        v2f b0, b1, b2, b3;
        b0.x = Wk[col];          b0.y = Wk[F + col];        // B[k+2g+v][col + 16t]
        b1.x = Wk[col + 16];     b1.y = Wk[F + col + 16];
        b2.x = Wk[col + 32];     b2.y = Wk[F + col + 32];
        b3.x = Wk[col + 48];     b3.y = Wk[F + col + 48];
        acc0 = __builtin_amdgcn_wmma_f32_16x16x4_f32(false, a, false, b0, (short)0, acc0, false, false);
        acc1 = __builtin_amdgcn_wmma_f32_16x16x4_f32(false, a, false, b1, (short)0, acc1, false, false);
        acc2 = __builtin_amdgcn_wmma_f32_16x16x4_f32(false, a, false, b2, (short)0, acc2, false, false);
        acc3 = __builtin_amdgcn_wmma_f32_16x16x4_f32(false, a, false, b3, (short)0, acc3, false, false);
    }

    float* Yt = Y + (size_t)rowbase * (size_t)F + colbase;
#pragma unroll
    for (int v = 0; v < 8; ++v) {
        size_t r = (size_t)(v + 8 * g) * (size_t)F;
        Yt[r + m]      = acc0[v];
        Yt[r + m + 16] = acc1[v];
        Yt[r + m + 32] = acc2[v];
        Yt[r + m + 48] = acc3[v];
    }
}

// ---------------- CSR gather: agg[d] += si[d] * sum_{e: dst(e)=d} so[src]*Y[src] ----------------
// One wave per destination node; contiguous float4 reads of Y rows; no float atomics.
template <int F>
__global__ void k_gather(const int* __restrict__ row_ptr, const int* __restrict__ col_idx,
                         const float* __restrict__ Y, const float* __restrict__ so,
                         const float* __restrict__ si, float* __restrict__ agg, int Nn) {
    int w    = (int)((blockIdx.x * blockDim.x + threadIdx.x) >> 5);
    int lane = threadIdx.x & 31;
    if (w >= Nn) return;
    int beg = row_ptr[w];
    int end = row_ptr[w + 1];
    float sid = si[w];
    float4 acc0 = make_float4(0.f, 0.f, 0.f, 0.f);
    float4 acc1 = make_float4(0.f, 0.f, 0.f, 0.f);
    for (int j = beg; j < end; ++j) {
        int s = col_idx[j];
        float c = so[s];
        const float4* Ys = (const float4*)(Y + (size_t)s * F);
        float4 y = Ys[lane];
        acc0.x += c * y.x; acc0.y += c * y.y; acc0.z += c * y.z; acc0.w += c * y.w;
        if (F == 256) {
            float4 y1 = Ys[lane + 32];
            acc1.x += c * y1.x; acc1.y += c * y1.y; acc1.z += c * y1.z; acc1.w += c * y1.w;
        }
    }
    float4* ad = (float4*)(agg + (size_t)w * F);
    float4 a = ad[lane];
    a.x += sid * acc0.x; a.y += sid * acc0.y; a.z += sid * acc0.z; a.w += sid * acc0.w;
    ad[lane] = a;
    if (F == 256) {
        float4 b = ad[lane + 32];
        b.x += sid * acc1.x; b.y += sid * acc1.y; b.z += sid * acc1.z; b.w += sid * acc1.w;
        ad[lane + 32] = b;
    }
}

// ---------------- semantic attention ----------------
__global__ void k_attn(const float* __restrict__ z, const float* __restrict__ Wp,
                       const float* __restrict__ bp, const float* __restrict__ q,
                       float* __restrict__ wsum_slot) {
    __shared__ float zr[OUTF];
    __shared__ float red[OUTF];
    int n = blockIdx.x;
    int j = threadIdx.x;                      // 128 threads
    zr[j] = z[(size_t)n * OUTF + j];
    __syncthreads();
    float s = bp[j];
#pragma unroll 8
    for (int i = 0; i < OUTF; ++i) s += zr[i] * Wp[i * OUTF + j];
    red[j] = tanhf(s) * q[j];
    __syncthreads();
    for (int off = OUTF >> 1; off > 0; off >>= 1) {
        if (j < off) red[j] += red[j + off];
        __syncthreads();
    }
    if (j == 0) atomicAdd(wsum_slot, red[0]);
}

__global__ void k_beta(const float* __restrict__ wsum, float* __restrict__ out_beta) {
    if (threadIdx.x != 0 || blockIdx.x != 0) return;
    for (int t = 0; t < 2; ++t) {
        float m0 = wsum[t * 3 + 0] / (float)NODES;
        float m1 = wsum[t * 3 + 1] / (float)NODES;
        float m2 = wsum[t * 3 + 2] / (float)NODES;
        float mx = fmaxf(m0, fmaxf(m1, m2));
        float e0 = expf(m0 - mx), e1 = expf(m1 - mx), e2 = expf(m2 - mx);
        float is = 1.0f / (e0 + e1 + e2);
        out_beta[t * 3 + 0] = e0 * is;
        out_beta[t * 3 + 1] = e1 * is;
        out_beta[t * 3 + 2] = e2 * is;
    }
}

__global__ void k_emb(const float* __restrict__ z0, const float* __restrict__ z1,
                      const float* __restrict__ z2, const float* __restrict__ beta,
                      float* __restrict__ emb, int total) {
    int t = blockIdx.x * blockDim.x + threadIdx.x;
    if (t >= total) return;
    emb[t] = beta[0] * z0[t] + beta[1] * z1[t] + beta[2] * z2[t];
}

// ---------------- host orchestration ----------------

extern "C" void kernel_launch(void* const* d_in, const int* in_sizes, int n_in,
                              void* d_out, int out_size, void* d_ws, size_t ws_size,
                              hipStream_t stream) {
    const float* xd  = (const float*)d_in[0];
    const float* xp  = (const float*)d_in[1];
    const int*   src = (const int*)  d_in[2];   // [4,4,E]
    const int*   dst = (const int*)  d_in[3];
    const float* W1  = (const float*)d_in[4];   // [4,4,512,256]
    const float* b1  = (const float*)d_in[5];   // [4,4,256]
    const float* W2  = (const float*)d_in[6];   // [4,4,256,128]
    const float* b2  = (const float*)d_in[7];   // [4,4,128]
    const float* Wp  = (const float*)d_in[8];
    const float* bp  = (const float*)d_in[9];
    const float* q   = (const float*)d_in[10];
    float* out = (float*)d_out;

    const size_t Nn = NODES;
    float* ws = (float*)d_ws;
    float* scl_out = ws;                                  // [3 slots][4 rel][N]
    float* scl_in  = scl_out + 12 * Nn;                   // [3][4][N]
    float* Yb      = scl_in  + 12 * Nn;                   // N*256 (reused for layer2)
    float* h1d     = Yb  + Nn * HIDF;                     // N*256
    float* h1p     = h1d + Nn * HIDF;                     // N*256
    float* zbuf    = h1p + Nn * HIDF;                     // 6 * N*128  (zd0..2, zp0..2)
    float* wsum    = zbuf + 6 * Nn * OUTF;                // 8 floats
    // integer region: CSR structures (built once, reused by both layers)
    int* row_ptr_all = (int*)(wsum + 8);                  // 12 * (N+1)
    int* col_idx_all = row_ptr_all + 12 * (NODES + 1);    // 12 * E
    int* cnt         = col_idx_all + (size_t)12 * NEDGE;  // N (doubles as cursor)

    const int chans[3] = {0, 2, 3};                       // reference stacks channels 0,2,3

    // ---- degree scales for used (channel, relation) pairs ----
    {
        int n = (int)(24 * Nn);
        k_zero<<<(n + 255) / 256, 256, 0, stream>>>(scl_out, n);
        for (int slot = 0; slot < 3; ++slot) {
            int c = chans[slot];
            for (int r = 0; r < 4; ++r) {
                const int* sp = src + (size_t)(c * 4 + r) * NEDGE;
                const int* dp = dst + (size_t)(c * 4 + r) * NEDGE;
                k_count_deg<<<(NEDGE + 255) / 256, 256, 0, stream>>>(
                    sp, dp, scl_out + (size_t)(slot * 4 + r) * Nn,
                            scl_in  + (size_t)(slot * 4 + r) * Nn, NEDGE);
            }
        }
        k_rsqrt_clamp<<<(n + 255) / 256, 256, 0, stream>>>(scl_out, n);
    }

    // ---- build CSR (dst-sorted adjacency) per (slot, relation) ----
    for (int slot = 0; slot < 3; ++slot) {
        int c = chans[slot];
        for (int r = 0; r < 4; ++r) {
            int idx = slot * 4 + r;
            const int* sp = src + (size_t)(c * 4 + r) * NEDGE;
            const int* dp = dst + (size_t)(c * 4 + r) * NEDGE;
            int* rp = row_ptr_all + (size_t)idx * (NODES + 1);
            int* ci = col_idx_all + (size_t)idx * NEDGE;
            k_zero_int<<<(NODES + 255) / 256, 256, 0, stream>>>(cnt, NODES);
            k_count_int<<<(NEDGE + 255) / 256, 256, 0, stream>>>(dp, cnt, NEDGE);
            k_scan<<<1, 1024, 0, stream>>>(cnt, rp, NODES);
            k_zero_int<<<(NODES + 255) / 256, 256, 0, stream>>>(cnt, NODES); // cursor
            k_fill<<<(NEDGE + 255) / 256, 256, 0, stream>>>(sp, dp, rp, cnt, ci, NEDGE);
        }
    }

    const int row_tiles = NODES / 16;                     // 1875 exactly
    const int gw_blocks = (NODES * 32 + 255) / 256;       // wave-per-dst-node

    for (int slot = 0; slot < 3; ++slot) {
        int c = chans[slot];
        // ---- layer 1 ----
        int n1 = NODES * HIDF;
        k_init_bias2<<<(n1 + 255) / 256, 256, 0, stream>>>(
            h1d, b1 + (size_t)(c * 4 + 0) * HIDF, b1 + (size_t)(c * 4 + 2) * HIDF, HIDF - 1, n1);
        k_init_bias2<<<(n1 + 255) / 256, 256, 0, stream>>>(
            h1p, b1 + (size_t)(c * 4 + 1) * HIDF, b1 + (size_t)(c * 4 + 3) * HIDF, HIDF - 1, n1);
        for (int r = 0; r < 4; ++r) {
            const float* X = (r < 2) ? xd : xp;
            int strips = HIDF / 64;                       // 4
            int tasks = row_tiles * strips;
            int idx = slot * 4 + r;
            k_gemm_wmma_f32<<<(tasks + 7) / 8, 256, 0, stream>>>(
                X, W1 + (size_t)(c * 4 + r) * INF_ * HIDF, Yb, INF_, HIDF, row_tiles, strips);
            k_gather<HIDF><<<gw_blocks, 256, 0, stream>>>(
                row_ptr_all + (size_t)idx * (NODES + 1), col_idx_all + (size_t)idx * NEDGE,
                Yb, scl_out + (size_t)idx * Nn, scl_in + (size_t)idx * Nn,
                ((r & 1) == 0) ? h1d : h1p, NODES);
        }
        k_relu<<<(n1 + 255) / 256, 256, 0, stream>>>(h1d, n1);
        k_relu<<<(n1 + 255) / 256, 256, 0, stream>>>(h1p, n1);

        // ---- layer 2 (accumulate directly into z slots) ----
        float* zd_s = zbuf + (size_t)slot * Nn * OUTF;
        float* zp_s = zbuf + (size_t)(3 + slot) * Nn * OUTF;
        int n2 = NODES * OUTF;
        k_init_bias2<<<(n2 + 255) / 256, 256, 0, stream>>>(
            zd_s, b2 + (size_t)(c * 4 + 0) * OUTF, b2 + (size_t)(c * 4 + 2) * OUTF, OUTF - 1, n2);
        k_init_bias2<<<(n2 + 255) / 256, 256, 0, stream>>>(
            zp_s, b2 + (size_t)(c * 4 + 1) * OUTF, b2 + (size_t)(c * 4 + 3) * OUTF, OUTF - 1, n2);
        for (int r = 0; r < 4; ++r) {
            const float* X = (r < 2) ? h1d : h1p;
            int strips = OUTF / 64;                       // 2
            int tasks = row_tiles * strips;
            int idx = slot * 4 + r;
            k_gemm_wmma_f32<<<(tasks + 7) / 8, 256, 0, stream>>>(
                X, W2 + (size_t)(c * 4 + r) * HIDF * OUTF, Yb, HIDF, OUTF, row_tiles, strips);
            k_gather<OUTF><<<gw_blocks, 256, 0, stream>>>(
                row_ptr_all + (size_t)idx * (NODES + 1), col_idx_all + (size_t)idx * NEDGE,
                Yb, scl_out + (size_t)idx * Nn, scl_in + (size_t)idx * Nn,
                ((r & 1) == 0) ? zd_s : zp_s, NODES);
        }
        k_relu<<<(n2 + 255) / 256, 256, 0, stream>>>(zd_s, n2);
        k_relu<<<(n2 + 255) / 256, 256, 0, stream>>>(zp_s, n2);
    }

    // ---- semantic attention ----
    k_zero<<<1, 32, 0, stream>>>(wsum, 8);
    for (int slot = 0; slot < 3; ++slot) {
        k_attn<<<NODES, OUTF, 0, stream>>>(zbuf + (size_t)slot * Nn * OUTF, Wp, bp, q, wsum + slot);
        k_attn<<<NODES, OUTF, 0, stream>>>(zbuf + (size_t)(3 + slot) * Nn * OUTF, Wp, bp, q, wsum + 3 + slot);
    }
    float* beta_out = out + 2 * Nn * OUTF;                // [beta_drug(3), beta_pro(3)]
    k_beta<<<1, 32, 0, stream>>>(wsum, beta_out);

    int ntot = NODES * OUTF;
    k_emb<<<(ntot + 255) / 256, 256, 0, stream>>>(
        zbuf + 0 * Nn * OUTF, zbuf + 1 * Nn * OUTF, zbuf + 2 * Nn * OUTF,
        beta_out, out, ntot);
    k_emb<<<(ntot + 255) / 256, 256, 0, stream>>>(
        zbuf + 3 * Nn * OUTF, zbuf + 4 * Nn * OUTF, zbuf + 5 * Nn * OUTF,
        beta_out + 3, out + Nn * OUTF, ntot);
}